// ReplaceCLS10Percent_HyperlinkPredictionHead_37735582663219
// MI455X (gfx1250) — compile-verified
//
#include <hip/hip_runtime.h>
#include <hip/hip_bf16.h>

// ---------------------------------------------------------------------------
// Adaptive log-softmax head for MI455X / gfx1250 (wave32, fp32 WMMA).
//
// GEMM: block = 256 threads = 8 waves stacked in M; block tile = 512(M) x 64(N)
//       wave tile = 64(M) x 64(N) via 4x4 grid of v_wmma_f32_16x16x4_f32
//       accumulators (128 acc VGPRs, launch_bounds(256,1) -> no spills),
//       K stepped by 4, unrolled x2 (8 loads : 16 WMMAs per step).
// W matrices re-read only M/512 = 4 times.  Row-wise sum(exp) partials are
// produced with in-wave shfl reductions and combined in fixed order
// -> deterministic, no global float atomics, no LDS in the hot path.
// ---------------------------------------------------------------------------

typedef __attribute__((ext_vector_type(2))) float v2f;
typedef __attribute__((ext_vector_type(8))) float v8f;

#define NROWS 2048      // N rows
#define DIM   1024      // D
#define SEQ   512       // S
#define SHORTLIST 4000
#define CUT1  20000
#define NHEAD 4002      // shortlist + 2 cluster logits
#define NT0   16000
#define NT1   30000
#define KT0   256
#define KT1   64

// ---------------------------------------------------------------------------
// 1) gather: final_emb[r] = replace ? cls[b] : lhs[b, s]
// ---------------------------------------------------------------------------
__global__ void gather_kernel(const float* __restrict__ lhs,
                              const float* __restrict__ cls,
                              const int* __restrict__ bids,
                              const int* __restrict__ tids,
                              const unsigned char* __restrict__ mask,
                              float* __restrict__ emb) {
    int r = blockIdx.x;
    int b = bids[r];
    int s = tids[r];
    const float* src = mask[r] ? (cls + (size_t)b * DIM)
                               : (lhs + ((size_t)b * SEQ + s) * DIM);
    const float4* s4 = (const float4*)src;
    float4* d4 = (float4*)(emb + (size_t)r * DIM);
    d4[threadIdx.x] = s4[threadIdx.x];
}

// ---------------------------------------------------------------------------
// 2) WMMA fp32 GEMM:  C[M,Ncols] = X[M,K] @ W[Ncols,K]^T  (+bias)
//    MODE 0: store C.  MODE 1: per-row sum(exp(C+bias)) -> partial[M, gridDim.x]
// ---------------------------------------------------------------------------
template <int MODE, bool HAS_BIAS>
__global__ __launch_bounds__(256, 1)
void wmma_gemm_kernel(const float* __restrict__ X,
                      const float* __restrict__ W,
                      const float* __restrict__ bias,
                      float* __restrict__ out,
                      int K, int Ncols) {
    const int lane  = threadIdx.x & 31;
    const int wave  = threadIdx.x >> 5;
    const int nl    = lane & 15;
    const int gsel  = lane >> 4;                 // 0 or 1 (lane group)
    const int koff  = gsel << 1;                 // A/B fragment K split: 0 or 2
    const int mwave = blockIdx.y * 512 + wave * 64;
    const int nbase = blockIdx.x * 64;

    // A fragment rows (4 M-subtiles of 16)
    const float* xrow[4];
#pragma unroll
    for (int mi = 0; mi < 4; ++mi)
        xrow[mi] = X + (size_t)(mwave + mi * 16 + nl) * K + koff;

    // B fragment rows (4 N-subtiles of 16); clamp OOB columns (discarded later)
    const float* wrow[4];
#pragma unroll
    for (int sub = 0; sub < 4; ++sub) {
        int n = nbase + sub * 16 + nl;
        if (n > Ncols - 1) n = Ncols - 1;
        wrow[sub] = W + (size_t)n * K + koff;
    }

    v8f acc[4][4];
#pragma unroll
    for (int mi = 0; mi < 4; ++mi)
#pragma unroll
        for (int sub = 0; sub < 4; ++sub) acc[mi][sub] = v8f{};

#pragma unroll 2
    for (int k0 = 0; k0 < K; k0 += 4) {
        v2f a[4], b[4];
#pragma unroll
        for (int mi = 0; mi < 4; ++mi) a[mi] = *(const v2f*)(xrow[mi] + k0);
#pragma unroll
        for (int sub = 0; sub < 4; ++sub) b[sub] = *(const v2f*)(wrow[sub] + k0);
#pragma unroll
        for (int mi = 0; mi < 4; ++mi)
#pragma unroll
            for (int sub = 0; sub < 4; ++sub)
                acc[mi][sub] = __builtin_amdgcn_wmma_f32_16x16x4_f32(
                    false, a[mi], false, b[sub], (short)0, acc[mi][sub],
                    false, false);
    }

    // C/D layout: VGPR v, lanes 0-15 -> M=v, N=lane; lanes 16-31 -> M=v+8.
    if (MODE == 0) {
#pragma unroll
        for (int mi = 0; mi < 4; ++mi)
#pragma unroll
            for (int sub = 0; sub < 4; ++sub) {
                int n = nbase + sub * 16 + nl;
                if (n < Ncols) {
#pragma unroll
                    for (int v = 0; v < 8; ++v) {
                        int m = mwave + mi * 16 + v + (gsel << 3);
                        out[(size_t)m * Ncols + n] = acc[mi][sub][v];
                    }
                }
            }
    } else {
        // per-row sum(exp): each wave owns its 64 rows exclusively.
        // Reduce across the 16 lanes of each group with shfl (deterministic).
#pragma unroll
        for (int mi = 0; mi < 4; ++mi) {
#pragma unroll
            for (int v = 0; v < 8; ++v) {
                float s = 0.0f;
#pragma unroll
                for (int sub = 0; sub < 4; ++sub) {
                    int n = nbase + sub * 16 + nl;
                    if (n < Ncols) {
                        float val = acc[mi][sub][v];
                        if (HAS_BIAS) val += bias[n];
                        s += __expf(val);        // logits small: max-free LSE safe
                    }
                }
                s += __shfl_xor(s, 1, 32);
                s += __shfl_xor(s, 2, 32);
                s += __shfl_xor(s, 4, 32);
                s += __shfl_xor(s, 8, 32);
                if (nl == 0) {
                    int m = mwave + mi * 16 + v + (gsel << 3);
                    out[(size_t)m * gridDim.x + blockIdx.x] = s;
                }
            }
        }
    }
}

// ---------------------------------------------------------------------------
// 3) finalize: one wave per row.
// ---------------------------------------------------------------------------
__device__ __forceinline__ float wave_reduce(float v) {
#pragma unroll
    for (int off = 16; off > 0; off >>= 1) v += __shfl_xor(v, off, 32);
    return v;
}

__global__ void finalize_kernel(const float* __restrict__ emb,
                                const float* __restrict__ h0,
                                const float* __restrict__ h1,
                                const int* __restrict__ targets,
                                const float* __restrict__ head_w,
                                const float* __restrict__ head_b,
                                const float* __restrict__ t0_out,
                                const float* __restrict__ t1_out,
                                const float* __restrict__ pH,
                                const float* __restrict__ pT0,
                                const float* __restrict__ pT1,
                                float* __restrict__ out,
                                float* __restrict__ lossPart,
                                int gxH, int gxT0, int gxT1) {
    __shared__ float outs[8];
    const int lane = threadIdx.x & 31;
    const int wave = threadIdx.x >> 5;
    const int r = blockIdx.x * 8 + wave;

    const int t = targets[r];
    const float* e = emb + (size_t)r * DIM;

    // head dots: target column (if in shortlist) + both cluster columns
    const int th = (t < SHORTLIST) ? t : 0;
    const float* wt = head_w + (size_t)th * DIM;
    const float* w0 = head_w + (size_t)SHORTLIST * DIM;
    const float* w1 = head_w + (size_t)(SHORTLIST + 1) * DIM;
    float at = 0.0f, a0 = 0.0f, a1 = 0.0f;
    for (int k = lane; k < DIM; k += 32) {
        float ev = e[k];
        at += ev * wt[k];
        a0 += ev * w0[k];
        a1 += ev * w1[k];
    }
    at = wave_reduce(at);
    a0 = wave_reduce(a0);
    a1 = wave_reduce(a1);

    // tail target dot (wave-uniform branch)
    float atail = 0.0f;
    if (t >= SHORTLIST && t < CUT1) {
        const float* hr = h0 + (size_t)r * KT0;
        const float* wr = t0_out + (size_t)(t - SHORTLIST) * KT0;
        for (int k = lane; k < KT0; k += 32) atail += hr[k] * wr[k];
    } else if (t >= CUT1) {
        const float* hr = h1 + (size_t)r * KT1;
        const float* wr = t1_out + (size_t)(t - CUT1) * KT1;
        for (int k = lane; k < KT1; k += 32) atail += hr[k] * wr[k];
    }
    atail = wave_reduce(atail);

    // combine per-block LSE partials (fixed strided assignment + fixed
    // reduction tree -> deterministic)
    float sH = 0.0f, s0 = 0.0f, s1 = 0.0f;
    for (int x = lane; x < gxH; x += 32) sH += pH[(size_t)r * gxH + x];
    for (int x = lane; x < gxT0; x += 32) s0 += pT0[(size_t)r * gxT0 + x];
    for (int x = lane; x < gxT1; x += 32) s1 += pT1[(size_t)r * gxT1 + x];
    sH = wave_reduce(sH);
    s0 = wave_reduce(s0);
    s1 = wave_reduce(s1);

    if (lane == 0) {
        float lseH = logf(sH);
        float o;
        if (t < SHORTLIST) {
            o = (at + head_b[t]) - lseH;
        } else if (t < CUT1) {
            o = (a0 + head_b[SHORTLIST] - lseH) + atail - logf(s0);
        } else {
            o = (a1 + head_b[SHORTLIST + 1] - lseH) + atail - logf(s1);
        }
        out[r] = o;
        outs[wave] = o;
    }
    __syncthreads();
    if (threadIdx.x == 0) {
        float s = 0.0f;
        for (int i = 0; i < 8; ++i) s += outs[i];
        lossPart[blockIdx.x] = s;
    }
}

__global__ void loss_final_kernel(const float* __restrict__ lossPart,
                                  float* __restrict__ out, int nblocks) {
    if (threadIdx.x == 0) {
        float s = 0.0f;
        for (int i = 0; i < nblocks; ++i) s += lossPart[i];
        out[NROWS] = -s / (float)NROWS;
    }
}

// ---------------------------------------------------------------------------
// launcher
// ---------------------------------------------------------------------------
extern "C" void kernel_launch(void* const* d_in, const int* in_sizes, int n_in,
                              void* d_out, int out_size, void* d_ws, size_t ws_size,
                              hipStream_t stream) {
    const int*   targets = (const int*)d_in[0];
    const float* lhs     = (const float*)d_in[1];
    const float* cls     = (const float*)d_in[2];
    const int*   bids    = (const int*)d_in[3];
    const int*   tids    = (const int*)d_in[4];
    const unsigned char* rmask = (const unsigned char*)d_in[5];
    const float* head_w  = (const float*)d_in[6];
    const float* head_b  = (const float*)d_in[7];
    const float* t0_proj = (const float*)d_in[8];
    const float* t0_out  = (const float*)d_in[9];
    const float* t1_proj = (const float*)d_in[10];
    const float* t1_out  = (const float*)d_in[11];
    float* out = (float*)d_out;    // [2048 logprobs, 1 loss]

    // workspace carve-up (bytes)
    char* ws = (char*)d_ws;
    float* emb   = (float*)(ws);                 // 2048*1024*4 = 8,388,608
    float* h0    = (float*)(ws + 8388608);       // 2048*256*4  = 2,097,152
    float* h1    = (float*)(ws + 10485760);      // 2048*64*4   =   524,288
    float* pH    = (float*)(ws + 11010048);      // 2048*63*4   =   516,096
    float* pT0   = (float*)(ws + 11526144);      // 2048*250*4  = 2,048,000
    float* pT1   = (float*)(ws + 13574144);      // 2048*469*4  = 3,842,048
    float* lossP = (float*)(ws + 17416192);      // 256*4
    // total ~17.4 MB

    const int gxH  = (NHEAD + 63) / 64;   // 63
    const int gxT0 = (NT0   + 63) / 64;   // 250
    const int gxT1 = (NT1   + 63) / 64;   // 469
    const int mb   = NROWS / 512;         // 4

    // 1) gather / CLS replacement
    gather_kernel<<<NROWS, 256, 0, stream>>>(lhs, cls, bids, tids, rmask, emb);

    // 2) tail projections (store mode)
    wmma_gemm_kernel<0, false><<<dim3(KT0 / 64, mb), 256, 0, stream>>>(emb, t0_proj, nullptr, h0, DIM, KT0);
    wmma_gemm_kernel<0, false><<<dim3(KT1 / 64, mb), 256, 0, stream>>>(emb, t1_proj, nullptr, h1, DIM, KT1);

    // 3) GEMM + sum(exp) partials
    wmma_gemm_kernel<1, true ><<<dim3(gxH,  mb), 256, 0, stream>>>(emb, head_w, head_b, pH,  DIM, NHEAD);
    wmma_gemm_kernel<1, false><<<dim3(gxT0, mb), 256, 0, stream>>>(h0,  t0_out, nullptr, pT0, KT0, NT0);
    wmma_gemm_kernel<1, false><<<dim3(gxT1, mb), 256, 0, stream>>>(h1,  t1_out, nullptr, pT1, KT1, NT1);

    // 4) per-row assembly + loss partials, 5) loss
    finalize_kernel<<<NROWS / 8, 256, 0, stream>>>(emb, h0, h1, targets, head_w, head_b,
                                                   t0_out, t1_out, pH, pT0, pT1,
                                                   out, lossP, gxH, gxT0, gxT1);
    loss_final_kernel<<<1, 32, 0, stream>>>(lossP, out, NROWS / 8);
}